// Histogram1D_48558900249124
// MI455X (gfx1250) — compile-verified
//
#include <hip/hip_runtime.h>

// Soft (triangular-kernel) histogram, B=32 rows x N=1M elems, 100 bins, row-normalized.
// Memory-bound: 128 MiB @ 23.3 TB/s ~= 5.8 us floor. Each element touches at most 2 bins.

#define B_ROWS 32
#define BINS 100
#define WAVES_PER_WG 8
#define WG_THREADS 256
#define WGS_PER_ROW 64

typedef __attribute__((ext_vector_type(2))) float v2f;
typedef __attribute__((ext_vector_type(8))) float v8f;

__global__ void zero_ws_kernel(float* __restrict__ ws, int n) {
    int i = blockIdx.x * blockDim.x + threadIdx.x;
    if (i < n) ws[i] = 0.0f;
}

__device__ __forceinline__ void lds_fadd(float* p, float v) {
    // Non-returning LDS float atomic -> ds_add_f32 (one DS op per contribution).
    __hip_atomic_fetch_add(p, v, __ATOMIC_RELAXED, __HIP_MEMORY_SCOPE_WORKGROUP);
}

// Per-lane privatized LDS histograms: hist[wave][bin][lane].
// Word index = bin*32 + lane  ->  distinct lanes always hit distinct banks
// (32*dBin == dLane mod 64 forces dLane==0), so ds_add_f32 is conflict-free
// and contention-free. 102.4 KB LDS/WG: only possible with CDNA5's 320 KB WGP LDS.
__global__ void __launch_bounds__(WG_THREADS)
hist_main_kernel(const float* __restrict__ x, float* __restrict__ partial, int N) {
    __shared__ float hist[WAVES_PER_WG][BINS][32];
    const int tid  = threadIdx.x;
    const int lane = tid & 31;
    const int wave = tid >> 5;

    for (int i = tid; i < WAVES_PER_WG * BINS * 32; i += WG_THREADS)
        ((float*)hist)[i] = 0.0f;
    __syncthreads();

    float* myhist = &hist[wave][0][lane];          // stride 32 floats per bin
    const int row = blockIdx.y;
    const float4* xr = (const float4*)(x + (size_t)row * (size_t)N);
    const int n4 = N >> 2;
    const int stride = gridDim.x * WG_THREADS;

    for (int i4 = blockIdx.x * WG_THREADS + tid; i4 < n4; i4 += stride) {
        float4 v = xr[i4];                         // global_load_b128, fully coalesced
        float e[4] = {v.x, v.y, v.z, v.w};
#pragma unroll
        for (int k = 0; k < 4; ++k) {
            float xs = e[k];
            int j = (int)(xs * 100.0f);
            j = j < 0 ? 0 : (j > 99 ? 99 : j);
            float t  = xs - ((float)j + 0.5f) * 0.01f;   // t in [-delta/2, delta/2]
            float at = fabsf(t);
            lds_fadd(myhist + j * 32, 0.01f - at);       // own bin: delta - |t|
            int nb = (t >= 0.0f) ? j + 1 : j - 1;        // neighbor bin: |t|
            if ((unsigned)nb < 100u)
                lds_fadd(myhist + nb * 32, at);
        }
    }
    __syncthreads();

    // Reduce 8 waves x 32 lanes per bin, fold into global partials.
    if (tid < BINS) {
        float s = 0.0f;
        for (int w = 0; w < WAVES_PER_WG; ++w) {
            const float4* hp = (const float4*)&hist[w][tid][0];  // 16B-aligned
#pragma unroll
            for (int l4 = 0; l4 < 8; ++l4) {
                float4 q = hp[l4];
                s += (q.x + q.y) + (q.z + q.w);
            }
        }
        __hip_atomic_fetch_add(&partial[row * BINS + tid], s,
                               __ATOMIC_RELAXED, __HIP_MEMORY_SCOPE_AGENT);
    }
}

// Row-sum of the [32,100] counts via V_WMMA_F32_16X16X4_F32, then normalize.
// A = 16x4 tile of counts, B = ones in column 0 => D[m][0] = row sum (exact f32 FMA tree).
// A-frag: lane l holds (m = l%16, k = 2*(l/16)+{0,1}).
// D: VGPR v holds M = v + 8*(l/16) at N = l%16 -> lanes 0 and 16 carry the N=0 sums.
__global__ void __launch_bounds__(32)
normalize_wmma_kernel(const float* __restrict__ P, float* __restrict__ out) {
    __shared__ float rowsum[B_ROWS];
    const int lane  = threadIdx.x;      // single wave32, EXEC all-ones at WMMA
    const int m     = lane & 15;
    const int khalf = lane >> 4;

    v2f bfrag;
    float bval = (m == 0) ? 1.0f : 0.0f;   // B[k][0] = 1, all K rows identical
    bfrag.x = bval; bfrag.y = bval;

    for (int h = 0; h < 2; ++h) {          // rows 0..15, then 16..31
        const float* Prow = P + (h * 16 + m) * BINS;
        v8f acc = {0.f, 0.f, 0.f, 0.f, 0.f, 0.f, 0.f, 0.f};
#pragma unroll
        for (int kc = 0; kc < 25; ++kc) {  // 25 * K4 = 100 bins
            v2f a;
            a.x = Prow[kc * 4 + khalf * 2 + 0];
            a.y = Prow[kc * 4 + khalf * 2 + 1];
            acc = __builtin_amdgcn_wmma_f32_16x16x4_f32(
                false, a, false, bfrag, (short)0, acc, false, false);
        }
        if (m == 0) {                      // lanes 0 and 16 hold column N=0
#pragma unroll
            for (int v = 0; v < 8; ++v)
                rowsum[h * 16 + khalf * 8 + v] = acc[v];
        }
    }
    __syncthreads();

    for (int idx = lane; idx < B_ROWS * BINS; idx += 32) {
        int b = idx / BINS;
        out[idx] = P[idx] / (rowsum[b] + 1e-5f);
    }
}

extern "C" void kernel_launch(void* const* d_in, const int* in_sizes, int n_in,
                              void* d_out, int out_size, void* d_ws, size_t ws_size,
                              hipStream_t stream) {
    (void)n_in; (void)out_size; (void)ws_size;
    const float* x = (const float*)d_in[0];
    float* out = (float*)d_out;
    float* ws  = (float*)d_ws;                 // [32][100] f32 partial counts
    const int N = in_sizes[0] / B_ROWS;

    zero_ws_kernel<<<dim3((B_ROWS * BINS + 255) / 256), dim3(256), 0, stream>>>(
        ws, B_ROWS * BINS);

    dim3 grid(WGS_PER_ROW, B_ROWS);
    hist_main_kernel<<<grid, dim3(WG_THREADS), 0, stream>>>(x, ws, N);

    normalize_wmma_kernel<<<dim3(1), dim3(32), 0, stream>>>(ws, out);
}